// NeuralNetwork_84653805404400
// MI455X (gfx1250) — compile-verified
//
#include <hip/hip_runtime.h>
#include <cstddef>

#define H 8192
#define OUT_N 8
#define TB 1024                 // threads per hidden block (32 waves)
#define NWAVE (TB / 32)         // 32 waves
#define KC (H / NWAVE)          // 256 K-values per wave
#define NITER (KC / 4)          // 64 WMMA steps per wave

typedef __attribute__((ext_vector_type(2))) float v2f;
typedef __attribute__((ext_vector_type(8))) float v8f;

__device__ __forceinline__ float sigmoidf(float v) {
    return 1.0f / (1.0f + __expf(-v));
}

// ---------------------------------------------------------------------------
// Layer 1: act[n] = sigmoid( sum_{k<10} in[k] * W_in[k][n] + b0[n] )
// ---------------------------------------------------------------------------
__global__ void __launch_bounds__(256)
layer1_kernel(const float* __restrict__ x, const float* __restrict__ s,
              const float* __restrict__ W_in, const float* __restrict__ b0,
              float* __restrict__ act) {
    int n = blockIdx.x * blockDim.x + threadIdx.x;
    float in[10];
#pragma unroll
    for (int k = 0; k < 4; ++k) in[k] = x[k];          // wave-uniform -> SMEM
#pragma unroll
    for (int k = 0; k < 6; ++k) in[4 + k] = s[k];
    float acc = b0[n];
#pragma unroll
    for (int k = 0; k < 10; ++k) acc = fmaf(in[k], W_in[k * H + n], acc);
    act[n] = sigmoidf(acc);
}

// ---------------------------------------------------------------------------
// Hidden layer GEMV on the WMMA pipe: act_out = sigmoid(act_in @ W + bias)
//   W row-major [H][H] (k-major, n contiguous).
//   Block = 1024 threads (32 waves), owns a 32-column tile (two 16-col WMMA
//   tiles per wave). Wave w reduces K slab [w*256, w*256+256).
//   A (16x4 f32), only row M=0 live (branch-free lane-mask construction):
//     lane0: V0=a[k+0] V1=a[k+1]; lane16: V0=a[k+2] V1=a[k+3]; else 0.
//   B (4x16 f32): lanes 0-15 rows K+0/K+1, lanes 16-31 rows K+2/K+3.
//   Two B tiles (cols n0.. and n0+16..) share one A -> full 128B lines.
//   B loads are non-temporal (single-use weights) and software-pipelined.
// ---------------------------------------------------------------------------
__global__ void __launch_bounds__(TB)
hidden_kernel(const float* __restrict__ act_in, const float* __restrict__ W,
              const float* __restrict__ bias, float* __restrict__ act_out) {
    __shared__ __align__(16) float4 lds_act4[H / 4];   // 32 KB activation
    __shared__ float lds_part[NWAVE * 32];             // 4 KB partial rows

    const int lane  = threadIdx.x & 31;
    const int wv    = threadIdx.x >> 5;
    const int n0    = blockIdx.x * 32;
    const int kBase = wv * KC;

    const int  col    = n0 + (lane & 15);
    const int  rowOff = (lane < 16) ? 0 : 2;
    const float m0    = (lane == 0)  ? 1.0f : 0.0f;
    const float m16   = (lane == 16) ? 1.0f : 0.0f;

    const float* wp0 = W + (size_t)(kBase + rowOff) * H + col;        // tile 0
    const float* wp1 = wp0 + 16;                                       // tile 1

    // ---- pipeline prologue: first B tiles in flight before LDS barrier ----
    v2f b0, b1;
    b0.x = __builtin_nontemporal_load(wp0);
    b0.y = __builtin_nontemporal_load(wp0 + H);
    b1.x = __builtin_nontemporal_load(wp1);
    b1.y = __builtin_nontemporal_load(wp1 + H);

    // cooperative stage of the activation vector (reused by all 32 waves)
    {
        const float4* g4 = (const float4*)act_in;
        for (int i = threadIdx.x; i < H / 4; i += TB) lds_act4[i] = g4[i];
    }
    __syncthreads();

    v8f c0 = {0.f, 0.f, 0.f, 0.f, 0.f, 0.f, 0.f, 0.f};
    v8f c1 = {0.f, 0.f, 0.f, 0.f, 0.f, 0.f, 0.f, 0.f};

#pragma unroll 3
    for (int it = 0; it < NITER - 1; ++it) {
        wp0 += (size_t)4 * H;
        wp1 += (size_t)4 * H;
        v2f nb0, nb1;                              // prefetch next K-step
        nb0.x = __builtin_nontemporal_load(wp0);
        nb0.y = __builtin_nontemporal_load(wp0 + H);
        nb1.x = __builtin_nontemporal_load(wp1);
        nb1.y = __builtin_nontemporal_load(wp1 + H);

        float4 av = lds_act4[(kBase >> 2) + it];   // uniform broadcast read
        v2f A;
        A.x = fmaf(av.x, m0, av.z * m16);
        A.y = fmaf(av.y, m0, av.w * m16);

        // (neg_a, A, neg_b, B, c_mod, C, reuse_a, reuse_b)
        c0 = __builtin_amdgcn_wmma_f32_16x16x4_f32(false, A, false, b0,
                                                   (short)0, c0, false, false);
        c1 = __builtin_amdgcn_wmma_f32_16x16x4_f32(false, A, false, b1,
                                                   (short)0, c1, false, false);
        b0 = nb0;
        b1 = nb1;
    }
    {   // epilogue: last K-step, no prefetch
        float4 av = lds_act4[(kBase >> 2) + (NITER - 1)];
        v2f A;
        A.x = fmaf(av.x, m0, av.z * m16);
        A.y = fmaf(av.y, m0, av.w * m16);
        c0 = __builtin_amdgcn_wmma_f32_16x16x4_f32(false, A, false, b0,
                                                   (short)0, c0, false, false);
        c1 = __builtin_amdgcn_wmma_f32_16x16x4_f32(false, A, false, b1,
                                                   (short)0, c1, false, false);
    }

    // D row M=0 lives in VGPR0 (c[0]) of lanes 0..15 (N = lane)
    if (lane < 16) {
        lds_part[wv * 32 + lane]      = c0[0];
        lds_part[wv * 32 + 16 + lane] = c1[0];
    }
    __syncthreads();

    if (threadIdx.x < 32) {
        int n = n0 + threadIdx.x;
        float acc = bias[n];
#pragma unroll
        for (int w = 0; w < NWAVE; ++w) acc += lds_part[w * 32 + threadIdx.x];
        act_out[n] = sigmoidf(acc);
    }
}

// ---------------------------------------------------------------------------
// Output layer: out[o] = sum_k act[k] * W_out[k][o]  (no bias, no activation)
// ---------------------------------------------------------------------------
__global__ void __launch_bounds__(256)
out_kernel(const float* __restrict__ act, const float* __restrict__ W_out,
           float* __restrict__ out) {
    __shared__ float red[256 * OUT_N];
    float acc[OUT_N];
#pragma unroll
    for (int o = 0; o < OUT_N; ++o) acc[o] = 0.0f;

    for (int k = threadIdx.x; k < H; k += 256) {
        float a = act[k];
        const float* wr = W_out + (size_t)k * OUT_N;
#pragma unroll
        for (int o = 0; o < OUT_N; ++o) acc[o] = fmaf(a, wr[o], acc[o]);
    }
#pragma unroll
    for (int o = 0; o < OUT_N; ++o) red[threadIdx.x * OUT_N + o] = acc[o];
    __syncthreads();

    for (int stp = 128; stp > 0; stp >>= 1) {
        if (threadIdx.x < stp) {
#pragma unroll
            for (int o = 0; o < OUT_N; ++o)
                red[threadIdx.x * OUT_N + o] += red[(threadIdx.x + stp) * OUT_N + o];
        }
        __syncthreads();
    }
    if (threadIdx.x < OUT_N) out[threadIdx.x] = red[threadIdx.x];
}

// ---------------------------------------------------------------------------
// inputs (setup_inputs order): x[4], s[6], W_in[10*8192], W_hh[3*8192*8192],
//                              W_out[8192*8], b[5*8192]    (all float32)
// output: 8 float32
// ---------------------------------------------------------------------------
extern "C" void kernel_launch(void* const* d_in, const int* in_sizes, int n_in,
                              void* d_out, int out_size, void* d_ws, size_t ws_size,
                              hipStream_t stream) {
    (void)in_sizes; (void)n_in; (void)out_size;
    if (ws_size < 2 * (size_t)H * sizeof(float)) return;

    const float* x     = (const float*)d_in[0];
    const float* s     = (const float*)d_in[1];
    const float* W_in  = (const float*)d_in[2];
    const float* W_hh  = (const float*)d_in[3];
    const float* W_out = (const float*)d_in[4];
    const float* b     = (const float*)d_in[5];
    float* out  = (float*)d_out;
    float* actA = (float*)d_ws;
    float* actB = actA + H;

    layer1_kernel<<<H / 256, 256, 0, stream>>>(x, s, W_in, b /* b[0] */, actA);
    hidden_kernel<<<H / 32, TB, 0, stream>>>(actA, W_hh + (size_t)0 * H * H, b + 1 * H, actB);
    hidden_kernel<<<H / 32, TB, 0, stream>>>(actB, W_hh + (size_t)1 * H * H, b + 2 * H, actA);
    hidden_kernel<<<H / 32, TB, 0, stream>>>(actA, W_hh + (size_t)2 * H * H, b + 3 * H, actB);
    out_kernel<<<1, 256, 0, stream>>>(actB, W_out, out);
}